// KernelConv_18193481466024
// MI455X (gfx1250) — compile-verified
//
#include <hip/hip_runtime.h>

typedef float v4f __attribute__((ext_vector_type(4)));

#define B_   16
#define N_   8
#define C_   3
#define H_   48
#define W_   48
#define KK_  25
#define NK_  4
#define HW_  (H_ * W_)      // 2304
#define R_   8              // rows per block stripe
#define GR_  (W_ / 4)       // 12 float4 groups per row
#define TPB  (R_ * GR_)     // 96 threads = 3 wave32
#define PADW 52             // 48 + 2*2 halo
#define TROWS (R_ + 4)      // 12 padded rows

// Fused kernel:
//  Phase 0: async-stage the needed frame rows (3 channels) into LDS via
//           global_load_async_to_lds_b128 (ASYNCcnt path, no VGPR round-trip)
//  Phase A: build zero-padded channel-sum halo tile from staged LDS
//  Phase B: stream the 300 core planes once (NT, b128, 512B/wave coalesced):
//           pred_i[bn,i,h,w] = sum_k (0.25*sum_g core[g*75+k*3+i]*kw[g,i]) * fsum[k]
__global__ __launch_bounds__(TPB) void fused_kpn_kernel(
    const float* __restrict__ frames,
    const float* __restrict__ core,
    const float* __restrict__ kwp,
    float* __restrict__ out_i)
{
    __shared__ float stage[TROWS * C_ * W_];  // 12 rows x 3 ch x 48 = 6912 B
    __shared__ float tile[TROWS * PADW];      // padded channel-sum tile

    const int tid    = threadIdx.x;
    const int bn     = blockIdx.x / (H_ / R_);
    const int stripe = blockIdx.x % (H_ / R_);
    const int r0     = stripe * R_;

    const float* fb = frames + (size_t)bn * C_ * HW_;

    // ---- Phase 0: async global->LDS staging of valid frame rows ----
    for (int t = tid; t < TROWS * C_ * GR_; t += TPB) {   // 432 float4 transfers
        const int pr  = t / (C_ * GR_);
        const int rem = t - pr * (C_ * GR_);
        const int c   = rem / GR_;
        const int g   = rem - c * GR_;
        const int gr  = r0 + pr - 2;
        if (gr >= 0 && gr < H_) {
            const float* src = fb + c * HW_ + gr * W_ + g * 4;
            const unsigned dst =
                (unsigned)(size_t)(const void*)(stage + (pr * C_ + c) * W_ + g * 4);
            asm volatile("global_load_async_to_lds_b128 %0, %1, off"
                         :: "v"(dst), "v"(src) : "memory");
        }
    }
    asm volatile("s_wait_asynccnt 0x0" ::: "memory");
    __syncthreads();

    // ---- Phase A: zero-padded channel-sum tile (LDS -> LDS) ----
    for (int idx = tid; idx < TROWS * PADW; idx += TPB) {
        const int pr = idx / PADW, pc = idx % PADW;
        const int gr = r0 + pr - 2, gc = pc - 2;
        float v = 0.0f;
        if (gr >= 0 && gr < H_ && gc >= 0 && gc < W_) {
            const float* s = stage + pr * C_ * W_ + gc;
            v = s[0] + s[W_] + s[2 * W_];
        }
        tile[idx] = v;
    }
    __syncthreads();

    // ---- Phase B: each thread owns 4 consecutive pixels of one row ----
    const int lr  = tid / GR_;         // local row 0..7
    const int grp = tid - lr * GR_;    // float4 group 0..11
    const int row = r0 + lr;
    const int c0  = grp * 4;

    const float* cb = core + (size_t)bn * (KK_ * C_ * NK_) * HW_ + row * W_ + c0;
    const float* kb = kwp  + (size_t)bn * (NK_ * C_)       * HW_ + row * W_ + c0;

    // gfx1250 global_prefetch of the four g-slabs we are about to stream
    __builtin_prefetch(cb,              0, 1);
    __builtin_prefetch(cb +  75 * HW_,  0, 1);
    __builtin_prefetch(cb + 150 * HW_,  0, 1);
    __builtin_prefetch(cb + 225 * HW_,  0, 1);

    // kernel weights, scaled by 1/NK (folds the mean over groups); read-once -> NT
    v4f kwv[NK_][C_];
#pragma unroll
    for (int g = 0; g < NK_; ++g)
#pragma unroll
        for (int i = 0; i < C_; ++i) {
            v4f t = __builtin_nontemporal_load((const v4f*)(kb + (g * C_ + i) * HW_));
            kwv[g][i] = t * 0.25f;
        }

    v4f acc[C_] = { {0.f,0.f,0.f,0.f}, {0.f,0.f,0.f,0.f}, {0.f,0.f,0.f,0.f} };

#pragma unroll
    for (int ik = 0; ik < 5; ++ik) {
        // 8-wide sliding window of the channel-sum row (2 aligned b128 LDS reads)
        const v4f a = *(const v4f*)(tile + (lr + ik) * PADW + c0);
        const v4f b = *(const v4f*)(tile + (lr + ik) * PADW + c0 + 4);
        const float nbr[8] = { a.x, a.y, a.z, a.w, b.x, b.y, b.z, b.w };
#pragma unroll
        for (int jk = 0; jk < 5; ++jk) {
            const int k = ik * 5 + jk;
            // Batch all 12 loads of this k before any use -> deep LOADcnt window
            v4f am[NK_][C_];
#pragma unroll
            for (int g = 0; g < NK_; ++g)
#pragma unroll
                for (int i = 0; i < C_; ++i)
                    am[g][i] = __builtin_nontemporal_load(
                        (const v4f*)(cb + (size_t)(g * 75 + k * C_ + i) * HW_));

            const v4f fs = { nbr[jk], nbr[jk + 1], nbr[jk + 2], nbr[jk + 3] };
#pragma unroll
            for (int i = 0; i < C_; ++i) {
                const v4f w = am[0][i] * kwv[0][i] + am[1][i] * kwv[1][i]
                            + am[2][i] * kwv[2][i] + am[3][i] * kwv[3][i];
                acc[i] += w * fs;
            }
        }
    }

    float* ob = out_i + (size_t)bn * C_ * HW_ + row * W_ + c0;
#pragma unroll
    for (int i = 0; i < C_; ++i)
        *(v4f*)(ob + i * HW_) = acc[i];
}

// pred_img[b,i,h,w] = mean_n pred_img_i[b,n,i,h,w]  (vectorized float4)
__global__ __launch_bounds__(256) void mean_n_kernel(
    const float* __restrict__ pred_i, float* __restrict__ pred)
{
    const int per_b = C_ * HW_ / 4;                 // 1728 float4 per b
    const int idx = blockIdx.x * blockDim.x + threadIdx.x;
    if (idx >= B_ * per_b) return;
    const int b = idx / per_b;
    const int r = idx - b * per_b;
    const v4f* src = (const v4f*)pred_i;
    v4f s = {0.f, 0.f, 0.f, 0.f};
#pragma unroll
    for (int n = 0; n < N_; ++n)
        s += src[(size_t)(b * N_ + n) * per_b + r];
    ((v4f*)pred)[idx] = s * 0.125f;
}

extern "C" void kernel_launch(void* const* d_in, const int* in_sizes, int n_in,
                              void* d_out, int out_size, void* d_ws, size_t ws_size,
                              hipStream_t stream)
{
    const float* frames = (const float*)d_in[0];   // [16,8,3,48,48]
    const float* core   = (const float*)d_in[1];   // [16,8,4,25,3,48,48]
    const float* kwp    = (const float*)d_in[2];   // [16,8,4,3,48,48]

    float* pred_i = (float*)d_out;                              // 884736 floats
    float* pred   = (float*)d_out + (size_t)B_ * N_ * C_ * HW_; // +110592 floats

    const dim3 grid1(B_ * N_ * (H_ / R_));  // 768 blocks of 96 threads (3 waves)
    fused_kpn_kernel<<<grid1, TPB, 0, stream>>>(frames, core, kwp, pred_i);

    const int total4 = B_ * C_ * HW_ / 4;   // 27648
    mean_n_kernel<<<(total4 + 255) / 256, 256, 0, stream>>>(pred_i, pred);
}